// Brain_74388833566812
// MI455X (gfx1250) — compile-verified
//
#include <hip/hip_runtime.h>
#include <stdint.h>

#define N_NEURONS  4096
#define BATCH      2
#define STEPS      16
#define INPUT_SIZE 1024

#define THREADS    256
#define TILE_E     2048                 // edges per LDS tile (16KB as u64)
#define PAIRS_PT   (TILE_E / THREADS / 2)  // 4 x b128 async copies per thread

typedef unsigned int u32;
typedef unsigned long long u64;

__device__ __forceinline__ float act_tanh(float x) {
#if defined(__gfx1250__) && __has_builtin(__builtin_amdgcn_tanhf)
  return __builtin_amdgcn_tanhf(x);       // v_tanh_f32 (CDNA5 trans op)
#else
  return tanhf(x);
#endif
}

// ---------------------------------------------------------------- init state
__global__ void init_kernel(const float* __restrict__ input,
                            float* __restrict__ hist,   // [B][STEPS][N]
                            float* __restrict__ acc,    // [B][N]
                            u32* __restrict__ counter) {
  int i = blockIdx.x * blockDim.x + threadIdx.x;
  if (i < BATCH * N_NEURONS) {
    acc[i] = 0.0f;
    int b = i >> 12, n = i & (N_NEURONS - 1);
    hist[(size_t)b * STEPS * N_NEURONS + n] =
        (n < INPUT_SIZE) ? input[b * INPUT_SIZE + n] : 0.0f;
  }
  if (i == 0) *counter = 0u;
}

// ------------------------------------------------- compact live edges (d<16)
__global__ void compact_kernel(const int* __restrict__ conn,   // [2][E]
                               const int* __restrict__ delays, // [E]
                               const float* __restrict__ w,    // [E]
                               u64* __restrict__ edges,        // out packed
                               u32* __restrict__ counter, int E) {
  int stride = gridDim.x * blockDim.x;
  for (int i = blockIdx.x * blockDim.x + threadIdx.x; i < E; i += stride) {
    int d = delays[i];
    if ((unsigned)d < (unsigned)STEPS) {               // d in [0,15] can ever fire
      u32 pk = (u32)conn[i] | ((u32)conn[E + i] << 12) | ((u32)d << 24);
      u32 wb = __float_as_uint(w[i]);
      u32 pos = atomicAdd(counter, 1u);
      edges[pos] = ((u64)wb << 32) | (u64)pk;
    }
  }
}

// ------------------------------------------------------------- edge process
__device__ __forceinline__ void proc_edge(u64 e, bool inb, int t,
                                          const float* __restrict__ hist,
                                          float* sAcc) {
  if (!inb) return;
  u32 pk = (u32)e;
  float ww = __uint_as_float((u32)(e >> 32));
  int d = (pk >> 24) & 31;
  if (d < t) {                                  // valid = delay < t
    int row  = (d == 0) ? 0 : (t - d);          // history[-0] == history[0]
    int from = pk & 4095;
    int to   = (pk >> 12) & 4095;
    float x0 = hist[row * N_NEURONS + from];
    float x1 = hist[STEPS * N_NEURONS + row * N_NEURONS + from];
    atomicAdd(&sAcc[to], ww * x0);                       // ds_add_f32
    atomicAdd(&sAcc[N_NEURONS + to], ww * x1);
  }
}

// ------------------------- main scatter: async-to-LDS double-buffered tiles
__global__ void __launch_bounds__(THREADS) scatter_compact(
    const u64* __restrict__ edges, const u32* __restrict__ counter,
    const float* __restrict__ hist, float* __restrict__ acc, int t) {
  __shared__ float sAcc[BATCH * N_NEURONS];              // 32KB
  __shared__ __align__(16) u64 sEdge[2][TILE_E];         // 2 x 16KB staging

  for (int i = threadIdx.x; i < BATCH * N_NEURONS; i += THREADS) sAcc[i] = 0.0f;
  __syncthreads();

  const int n      = (int)*counter;
  const int ntiles = (n + TILE_E - 1) / TILE_E;
  const int tid    = threadIdx.x;
  const unsigned buf0 = (unsigned)(uintptr_t)&sEdge[0][0];  // LDS byte offsets
  const unsigned buf1 = (unsigned)(uintptr_t)&sEdge[1][0];

  auto stage = [&](int tileStart, unsigned ldsBuf) {
#pragma unroll
    for (int k = 0; k < PAIRS_PT; ++k) {
      int s = k * THREADS + tid;                          // pair slot 0..1023
      const u64* g = edges + tileStart + 2 * s;
      unsigned lds = ldsBuf + (unsigned)(s * 16);
      asm volatile("global_load_async_to_lds_b128 %0, %1, off"
                   :: "v"(lds), "v"(g) : "memory");
    }
  };

  int myTile = blockIdx.x;
  int cur = 0;
  if (myTile < ntiles) stage(myTile * TILE_E, buf0);
  while (myTile < ntiles) {
    int nxt = myTile + gridDim.x;
    bool hasNext = (nxt < ntiles);
    if (hasNext) stage(nxt * TILE_E, cur ? buf0 : buf1);
    if (hasNext) asm volatile("s_wait_asynccnt 4" ::: "memory");
    else         asm volatile("s_wait_asynccnt 0" ::: "memory");

    int base = myTile * TILE_E;
    const ulonglong2* tileVec = (const ulonglong2*)&sEdge[cur][0];
#pragma unroll
    for (int k = 0; k < PAIRS_PT; ++k) {
      int s  = k * THREADS + tid;
      ulonglong2 ee = tileVec[s];                 // ds_load_b128: both edges
      int g0 = base + 2 * s;
      proc_edge(ee.x, g0 + 0 < n, t, hist, sAcc);
      proc_edge(ee.y, g0 + 1 < n, t, hist, sAcc);
    }
    myTile += gridDim.x;
    cur ^= 1;
  }
  __syncthreads();
  for (int i = tid; i < BATCH * N_NEURONS; i += THREADS) {
    float v = sAcc[i];
    if (v != 0.0f) atomicAdd(&acc[i], v);
  }
}

// -------------------------------------- fallback: stream raw arrays (no ws)
__global__ void __launch_bounds__(THREADS) scatter_raw(
    const int* __restrict__ conn, const int* __restrict__ delays,
    const float* __restrict__ w, int E,
    const float* __restrict__ hist, float* __restrict__ acc, int t) {
  __shared__ float sAcc[BATCH * N_NEURONS];
  for (int i = threadIdx.x; i < BATCH * N_NEURONS; i += THREADS) sAcc[i] = 0.0f;
  __syncthreads();
  int stride = gridDim.x * blockDim.x;
  for (int i = blockIdx.x * blockDim.x + threadIdx.x; i < E; i += stride) {
    int d = delays[i];
    if (d < t) {
      int row  = (d == 0) ? 0 : (t - d);
      int from = conn[i], to = conn[E + i];
      float ww = w[i];
      float x0 = hist[row * N_NEURONS + from];
      float x1 = hist[STEPS * N_NEURONS + row * N_NEURONS + from];
      atomicAdd(&sAcc[to], ww * x0);
      atomicAdd(&sAcc[N_NEURONS + to], ww * x1);
    }
  }
  __syncthreads();
  for (int i = threadIdx.x; i < BATCH * N_NEURONS; i += THREADS) {
    float v = sAcc[i];
    if (v != 0.0f) atomicAdd(&acc[i], v);
  }
}

// ------------------------------------------------- tanh + history/out write
__global__ void update_kernel(float* __restrict__ hist, float* __restrict__ acc,
                              float* __restrict__ out, int t) {
  int i = blockIdx.x * blockDim.x + threadIdx.x;
  if (i >= BATCH * N_NEURONS) return;
  int b = i >> 12, nn = i & (N_NEURONS - 1);
  float v = act_tanh(acc[i]);
  acc[i] = 0.0f;
  if (t < STEPS) {
    hist[(size_t)b * STEPS * N_NEURONS + (size_t)t * N_NEURONS + nn] = v;
  } else if (nn >= N_NEURONS - INPUT_SIZE) {
    out[b * INPUT_SIZE + (nn - (N_NEURONS - INPUT_SIZE))] = v;
  }
}

// --------------------------------------------------------------------- host
extern "C" void kernel_launch(void* const* d_in, const int* in_sizes, int n_in,
                              void* d_out, int out_size, void* d_ws, size_t ws_size,
                              hipStream_t stream) {
  const float* input   = (const float*)d_in[0];
  const float* weights = (const float*)d_in[1];
  const int*   conn    = (const int*)d_in[2];
  const int*   delays  = (const int*)d_in[3];
  const int E = in_sizes[1];

  const size_t Epad     = ((size_t)E + TILE_E - 1) / TILE_E * TILE_E;
  const size_t histB    = (size_t)BATCH * STEPS * N_NEURONS * sizeof(float);
  const size_t accB     = (size_t)BATCH * N_NEURONS * sizeof(float);
  const size_t needA    = Epad * 8 + histB + accB + 256;
  const bool   compact  = (ws_size >= needA);

  char* ws = (char*)d_ws;
  u64*   edges = nullptr;
  float* hist; float* acc; u32* cnt;
  if (compact) {
    edges = (u64*)ws;
    hist  = (float*)(ws + Epad * 8);
    acc   = (float*)(ws + Epad * 8 + histB);
    cnt   = (u32*)(ws + Epad * 8 + histB + accB);
  } else {
    hist  = (float*)ws;
    acc   = (float*)(ws + histB);
    cnt   = (u32*)(ws + histB + accB);
  }

  init_kernel<<<(BATCH * N_NEURONS + 255) / 256, 256, 0, stream>>>(input, hist, acc, cnt);
  if (compact)
    compact_kernel<<<2048, 256, 0, stream>>>(conn, delays, weights, edges, cnt, E);

  for (int t = 1; t <= STEPS; ++t) {
    if (compact)
      scatter_compact<<<256, THREADS, 0, stream>>>(edges, cnt, hist, acc, t);
    else
      scatter_raw<<<512, THREADS, 0, stream>>>(conn, delays, weights, E, hist, acc, t);
    update_kernel<<<(BATCH * N_NEURONS + 255) / 256, 256, 0, stream>>>(
        hist, acc, (float*)d_out, t);
  }
}